// EncoderBlock_15101105013507
// MI455X (gfx1250) — compile-verified
//
#include <hip/hip_runtime.h>
#include <math.h>

#define B_  8
#define D_  128
#define L_  1024
#define K_  7
#define CONV_NUM_ 4
#define NH_ 8
#define DK_ 16

typedef __attribute__((ext_vector_type(2))) float v2f;
typedef __attribute__((ext_vector_type(8))) float v8f;

// D = A(16x4, f32) * B(4x16, f32) + C(16x16, f32) — exact fp32 matrix op on CDNA5
__device__ __forceinline__ v8f wmma4(v2f a, v2f b, v8f c) {
  return __builtin_amdgcn_wmma_f32_16x16x4_f32(false, a, false, b, (short)0, c,
                                               false, false);
}

// gfx1250 async global->LDS copy (ASYNCcnt-tracked), 4 bytes per lane.
__device__ __forceinline__ void async_ld_b32(uint32_t lds_off,
                                             const float* gptr) {
  asm volatile("global_load_async_to_lds_b32 %0, %1, off" ::"v"(lds_off),
               "v"(gptr)
               : "memory");
}

// ---------------------------------------------------------------- add pos enc
__global__ void add_pos_kernel(const float* __restrict__ x,
                               const float* __restrict__ pe,
                               float* __restrict__ out) {
  int idx = blockIdx.x * blockDim.x + threadIdx.x;
  const int n = B_ * D_ * L_;
  if (idx < n) {
    int dl = idx % (D_ * L_);
    out[idx] = x[idx] + pe[dl];
  }
}

// ---------------------------------------------------------------- LayerNorm over [D,L] per batch
__global__ void ln_kernel(const float* __restrict__ y,
                          const float* __restrict__ w,
                          const float* __restrict__ bb,
                          float* __restrict__ out) {
  const int b = blockIdx.x;
  const int n = D_ * L_;
  const float* yb = y + (size_t)b * n;
  float* ob = out + (size_t)b * n;
  __shared__ float ssum[256];
  __shared__ float ssq[256];
  float s = 0.f, q = 0.f;
  for (int i = threadIdx.x; i < n; i += 256) {
    float v = yb[i];
    s += v;
    q += v * v;
  }
  ssum[threadIdx.x] = s;
  ssq[threadIdx.x] = q;
  __syncthreads();
  for (int off = 128; off > 0; off >>= 1) {
    if (threadIdx.x < off) {
      ssum[threadIdx.x] += ssum[threadIdx.x + off];
      ssq[threadIdx.x] += ssq[threadIdx.x + off];
    }
    __syncthreads();
  }
  float mean = ssum[0] * (1.f / n);
  float var = ssq[0] * (1.f / n) - mean * mean;
  float rstd = rsqrtf(var + 1e-5f);
  for (int i = threadIdx.x; i < n; i += 256)
    ob[i] = (yb[i] - mean) * rstd * w[i] + bb[i];
}

// ---------------------------------------------------------------- depthwise conv1d K=7, pad 3
__global__ void dwconv_kernel(const float* __restrict__ y,
                              const float* __restrict__ dwk,
                              float* __restrict__ out) {
  int idx = blockIdx.x * blockDim.x + threadIdx.x;
  const int n = B_ * D_ * L_;
  if (idx >= n) return;
  int l = idx % L_;
  int bd = idx / L_;
  int d = bd % D_;
  const float* kr = dwk + d * K_;
  const float* row = y + (size_t)bd * L_;
  float acc = 0.f;
#pragma unroll
  for (int k = 0; k < K_; ++k) {
    int ll = l + k - (K_ / 2);
    if (ll >= 0 && ll < L_) acc = fmaf(kr[k], row[ll], acc);
  }
  out[idx] = acc;
}

// ================================================================ GEMM v2
// M = 128 batched GEMM with B staged in LDS via async-to-LDS, double buffered.
// Block = 8 waves; wave w computes the 16x64 tile at rows w*16, cols n0..n0+63.
// B slab (16 k-rows x 64 cols) stored pair-interleaved:
//   element (k, n) at [(k>>1)*BSTR + n*2 + (k&1)]  -> (k,k+1) pair = one b64.
// BSTR = 160 DWORDs => half-waves of a ds_load_b64 hit disjoint LDS banks.
#define BSTR 160

__global__ __launch_bounds__(256) void gemm_lds_kernel(
    const float* __restrict__ Wm, const float* __restrict__ X,
    float* __restrict__ Y, const float* __restrict__ resid, int Kd, int N,
    int relu) {
  __shared__ float shB[2][8 * BSTR];  // 2 stages x 5120B
  const int g = blockIdx.y;
  const int tid = threadIdx.x;
  const int lane = tid & 31;
  const int wid = tid >> 5;
  const int lane16 = lane & 15;
  const int hi = lane >> 4;
  const int m0 = wid << 4;
  const int n0 = blockIdx.x << 6;

  const float* Wg = Wm + (size_t)g * D_ * Kd;
  const float* Xg = X + (size_t)g * Kd * N;

  // issue stage 0 fill: 1024 elements, 4 async b32 per thread, coalesced in n
#define STAGE_FILL(BUF, KK)                                                  \
  {                                                                          \
    _Pragma("unroll") for (int i = 0; i < 4; ++i) {                          \
      int e = tid + i * 256;                                                 \
      int k = e >> 6, nn = e & 63;                                           \
      uint32_t lo = (uint32_t)(uintptr_t)(&shB[BUF][0] + (k >> 1) * BSTR +   \
                                          nn * 2 + (k & 1));                 \
      async_ld_b32(lo, Xg + (size_t)((KK) + k) * N + n0 + nn);               \
    }                                                                        \
  }

  STAGE_FILL(0, 0)

  v8f acc0 = {}, acc1 = {}, acc2 = {}, acc3 = {};

  for (int kk = 0; kk < Kd; kk += 16) {
    const int cb = (kk >> 4) & 1;
    if (kk + 16 < Kd) {
      STAGE_FILL(cb ^ 1, kk + 16)
      asm volatile("s_wait_asynccnt 0x4" ::: "memory");  // stage cb complete
    } else {
      asm volatile("s_wait_asynccnt 0x0" ::: "memory");
    }
    __syncthreads();  // all waves' fills of stage cb visible

    const float* bS = &shB[cb][0];
#pragma unroll
    for (int c = 0; c < 4; ++c) {
      const int kd = 4 * c + 2 * hi;  // local k; even -> aligned pair
      v2f av = *(const v2f*)(Wg + (size_t)(m0 + lane16) * Kd + kk + kd);
      const float* bp = bS + (kd >> 1) * BSTR + lane16 * 2;
#define BT(ACC, NOFF)                                     \
  {                                                       \
    v2f bv = *(const v2f*)(bp + (NOFF) * 2);              \
    ACC = wmma4(av, bv, ACC);                             \
  }
      BT(acc0, 0)
      BT(acc1, 16)
      BT(acc2, 32)
      BT(acc3, 48)
#undef BT
    }
    __syncthreads();  // reads of stage cb done before it is refilled
  }
#undef STAGE_FILL

  const size_t base = (size_t)g * D_ * N;
#define EPI(ACC, NOFF)                                        \
  {                                                           \
    const int nn = n0 + (NOFF) + lane16;                      \
    _Pragma("unroll") for (int r = 0; r < 8; ++r) {           \
      float vv = ACC[r];                                      \
      if (relu) vv = fmaxf(vv, 0.f);                          \
      size_t o = base + (size_t)(m0 + r + 8 * hi) * N + nn;   \
      if (resid) vv += resid[o];                              \
      Y[o] = vv;                                              \
    }                                                         \
  }
  EPI(acc0, 0)
  EPI(acc1, 16)
  EPI(acc2, 32)
  EPI(acc3, 48)
#undef EPI
}

// ================================================================ GEMM v1
// Generic wave-tile GEMM (used for the M=16 Q/K/V projections).
__global__ __launch_bounds__(256) void gemm_wmma_kernel(
    const float* __restrict__ Wm, const float* __restrict__ X,
    float* __restrict__ Y, const float* __restrict__ resid, int M, int Kd,
    int N, int in_mod, int relu) {
  const int g = blockIdx.y;
  const int lane = threadIdx.x & 31;
  const int wid = threadIdx.x >> 5;
  const int lane16 = lane & 15;
  const int hi = lane >> 4;
  const int mTiles = M >> 4;
  const int nBlocks = N >> 6;
  const int t = blockIdx.x * 8 + wid;
  if (t >= mTiles * nBlocks) return;
  const int m0 = (t % mTiles) << 4;
  const int n0 = (t / mTiles) << 6;

  const float* Wg = Wm + (size_t)g * M * Kd;
  const float* Xg = X + (size_t)(g % in_mod) * Kd * N;

  v8f acc0 = {}, acc1 = {}, acc2 = {}, acc3 = {};

  for (int kk = 0; kk < Kd; kk += 16) {
    if (kk + 16 < Kd)  // gfx1250 global_prefetch of next K slab
      __builtin_prefetch(Xg + (size_t)(kk + 16 + 2 * hi) * N + n0 + lane16, 0,
                         1);
#pragma unroll
    for (int c = 0; c < 4; ++c) {
      const int kd = kk + 4 * c + 2 * hi;
      v2f av = *(const v2f*)(Wg + (size_t)(m0 + lane16) * Kd + kd);
      const float* xb = Xg + (size_t)kd * N;
#define BTILE(ACC, NOFF)                                 \
  {                                                      \
    v2f bv;                                              \
    bv.x = xb[n0 + (NOFF) + lane16];                     \
    bv.y = xb[(size_t)N + n0 + (NOFF) + lane16];         \
    ACC = wmma4(av, bv, ACC);                            \
  }
      BTILE(acc0, 0)
      BTILE(acc1, 16)
      BTILE(acc2, 32)
      BTILE(acc3, 48)
#undef BTILE
    }
  }

  const size_t base = (size_t)g * M * N;
#define EPI(ACC, NOFF)                                        \
  {                                                           \
    const int nn = n0 + (NOFF) + lane16;                      \
    _Pragma("unroll") for (int r = 0; r < 8; ++r) {           \
      float vv = ACC[r];                                      \
      if (relu) vv = fmaxf(vv, 0.f);                          \
      size_t o = base + (size_t)(m0 + r + 8 * hi) * N + nn;   \
      if (resid) vv += resid[o];                              \
      Y[o] = vv;                                              \
    }                                                         \
  }
  EPI(acc0, 0)
  EPI(acc1, 16)
  EPI(acc2, 32)
  EPI(acc3, 48)
#undef EPI
}

// ---------------------------------------------------------------- flash attention
// scores[l,m] = sum_d Q[d,l]*K[d,m] * (1/sqrt(DK)); softmax over l (rows);
// head[v,m]  = sum_l V[v,l] * P[l,m].  One wave per 16 output columns m.
// P bounced through LDS in (l,l+1)-pair layout: pair stride 32 DWORDs ->
// b64 stores/loads, load half-waves on disjoint banks.
__global__ __launch_bounds__(256) void attn_kernel(const float* __restrict__ Q,
                                                   const float* __restrict__ Kt,
                                                   const float* __restrict__ V,
                                                   float* __restrict__ head) {
  __shared__ float sh[8 * 8 * 32];  // 8 waves x (8 pairs x 16 cols x 2)
  const int hb = blockIdx.y;        // h*B + b
  const int b = hb % B_;
  const int h = hb / B_;
  const int lane = threadIdx.x & 31;
  const int wid = threadIdx.x >> 5;
  const int lane16 = lane & 15;
  const int hi = lane >> 4;
  const int m0 = blockIdx.x * 128 + wid * 16;

  const float* Qg = Q + (size_t)hb * DK_ * L_;
  const float* Kg = Kt + (size_t)hb * DK_ * L_;
  const float* Vg = V + (size_t)hb * DK_ * L_;
  float* shp = sh + wid * (8 * 32);

  // Fixed B operand: K columns of this wave's m-tile. N=lane16, K(d)=vgpr+2*hi.
  v2f bk[4];
#pragma unroll
  for (int c = 0; c < 4; ++c) {
    const int d = 4 * c + 2 * hi;
    bk[c].x = Kg[(size_t)d * L_ + m0 + lane16];
    bk[c].y = Kg[(size_t)(d + 1) * L_ + m0 + lane16];
  }

  v8f hacc = {};
  float runmax = -3.0e38f, runsum = 0.f;

  for (int l0 = 0; l0 < L_; l0 += 16) {
    // S tile = (Q-tile)^T * K-tile : A rows = l', contraction over d (4 wmma)
    v8f s = {};
#pragma unroll
    for (int c = 0; c < 4; ++c) {
      const int d = 4 * c + 2 * hi;
      v2f av;
      av.x = Qg[(size_t)d * L_ + l0 + lane16];
      av.y = Qg[(size_t)(d + 1) * L_ + l0 + lane16];
      s = wmma4(av, bk[c], s);
    }
    // online softmax over l (column-wise in D layout: 8 VGPRs + lane^16)
    float tmax = -3.0e38f;
#pragma unroll
    for (int r = 0; r < 8; ++r) {
      s[r] *= 0.25f;  // 1/sqrt(DK)
      tmax = fmaxf(tmax, s[r]);
    }
    tmax = fmaxf(tmax, __shfl_xor(tmax, 16, 32));
    const float nmax = fmaxf(runmax, tmax);
    const float scale = __expf(runmax - nmax);
    float psum = 0.f;
#pragma unroll
    for (int r = 0; r < 8; ++r) {
      float p = __expf(s[r] - nmax);
      s[r] = p;
      psum += p;
    }
    psum += __shfl_xor(psum, 16, 32);
    runsum = runsum * scale + psum;
    runmax = nmax;
#pragma unroll
    for (int r = 0; r < 8; ++r) hacc[r] *= scale;

    // D-layout -> LDS pair layout: (l', l'+1) adjacent -> ds_store_b64
#pragma unroll
    for (int r = 0; r < 8; r += 2) {
      v2f pv;
      pv.x = s[r];
      pv.y = s[r + 1];
      *(v2f*)(shp + ((r + 8 * hi) >> 1) * 32 + lane16 * 2) = pv;
    }

    // hacc += V-tile (A: M=v, K=l') * P (B from LDS, one b64 per chunk)
#pragma unroll
    for (int c = 0; c < 4; ++c) {
      const int lofs = 4 * c + 2 * hi;
      v2f a2 = *(const v2f*)(Vg + (size_t)lane16 * L_ + l0 + lofs);
      v2f pb = *(const v2f*)(shp + (lofs >> 1) * 32 + lane16 * 2);
      hacc = wmma4(a2, pb, hacc);
    }
  }

  const float inv = 1.f / runsum;
  float* hg = head + ((size_t)b * D_ + (size_t)h * DK_) * L_;
#pragma unroll
  for (int r = 0; r < 8; ++r)
    hg[(size_t)(r + 8 * hi) * L_ + m0 + lane16] = hacc[r] * inv;
}

// ---------------------------------------------------------------- host orchestration
extern "C" void kernel_launch(void* const* d_in, const int* in_sizes, int n_in,
                              void* d_out, int out_size, void* d_ws,
                              size_t ws_size, hipStream_t stream) {
  const float* x   = (const float*)d_in[0];
  const float* pe  = (const float*)d_in[1];
  const float* lnw = (const float*)d_in[2];
  const float* lnb = (const float*)d_in[3];
  const float* dw  = (const float*)d_in[4];
  const float* pw  = (const float*)d_in[5];
  const float* Wq  = (const float*)d_in[6];
  const float* Wk  = (const float*)d_in[7];
  const float* Wv  = (const float*)d_in[8];
  const float* Wo  = (const float*)d_in[9];
  const float* Wf  = (const float*)d_in[10];
  float* out = (float*)d_out;

  const size_t SZ = (size_t)B_ * D_ * L_;  // 1M floats
  float* buf0 = (float*)d_ws;
  float* buf1 = buf0 + SZ;
  float* tmp  = buf0 + 2 * SZ;
  float* dwb  = buf0 + 3 * SZ;
  float* q    = buf0 + 4 * SZ;
  float* k    = buf0 + 5 * SZ;
  float* v    = buf0 + 6 * SZ;
  float* head = buf0 + 7 * SZ;

  const int n = B_ * D_ * L_;
  add_pos_kernel<<<(n + 255) / 256, 256, 0, stream>>>(x, pe, buf0);

  float* cur = buf0;
  float* nxt = buf1;
  for (int i = 0; i < CONV_NUM_; ++i) {
    ln_kernel<<<B_, 256, 0, stream>>>(cur, lnw, lnb, tmp);
    dwconv_kernel<<<(n + 255) / 256, 256, 0, stream>>>(
        tmp, dw + (size_t)i * D_ * K_, dwb);
    // pointwise 1x1 + ReLU + residual(cur): LDS-staged GEMM, grid(N/64, B)
    gemm_lds_kernel<<<dim3(L_ / 64, B_), 256, 0, stream>>>(
        pw + (size_t)i * D_ * D_, dwb, nxt, cur, D_, L_, 1);
    float* t2 = cur; cur = nxt; nxt = t2;
  }

  ln_kernel<<<B_, 256, 0, stream>>>(cur, lnw, lnb, tmp);
  // Q/K/V projections: G = NH*B, M=16 -> 16 wave-tiles per group -> grid(2, 64)
  gemm_wmma_kernel<<<dim3(2, NH_ * B_), 256, 0, stream>>>(
      Wq, tmp, q, nullptr, DK_, D_, L_, B_, 0);
  gemm_wmma_kernel<<<dim3(2, NH_ * B_), 256, 0, stream>>>(
      Wk, tmp, k, nullptr, DK_, D_, L_, B_, 0);
  gemm_wmma_kernel<<<dim3(2, NH_ * B_), 256, 0, stream>>>(
      Wv, tmp, v, nullptr, DK_, D_, L_, B_, 0);

  attn_kernel<<<dim3(L_ / 128, NH_ * B_), 256, 0, stream>>>(q, k, v, head);

  // Wo @ head + residual(cur) -> nxt
  gemm_lds_kernel<<<dim3(L_ / 64, B_), 256, 0, stream>>>(Wo, head, nxt, cur,
                                                         D_, L_, 0);
  { float* t2 = cur; cur = nxt; nxt = t2; }

  ln_kernel<<<B_, 256, 0, stream>>>(cur, lnw, lnb, tmp);
  // final: relu(W @ ln) + residual(cur) -> d_out
  gemm_lds_kernel<<<dim3(L_ / 64, B_), 256, 0, stream>>>(Wf, tmp, out, cur, D_,
                                                         L_, 1);
}